// ModelNew_25056839204927
// MI455X (gfx1250) — compile-verified
//
#include <hip/hip_runtime.h>

// Axial 7x7 dual attention, fused single pass. One workgroup per n.
// Memory-bound (~180 MB @ 23.3 TB/s => ~8us floor). Scores via
// V_WMMA_F32_16X16X4_F32 (f32 inputs, K=64 via 16 chained WMMAs), with all
// fragment loads hoisted (branch-free zero-padding) so loads clause together
// and the WMMA chain runs back-to-back instead of load-wait-wmma serialization.

#define G7   7
#define CQ   64
#define CV   512
#define NEGV (-1e20f)

typedef float v2f __attribute__((ext_vector_type(2)));
typedef float v8f __attribute__((ext_vector_type(8)));

__global__ __launch_bounds__(256) void axial_attn_kernel(
    const float* __restrict__ qH, const float* __restrict__ kH,
    const float* __restrict__ vH, const float* __restrict__ qW,
    const float* __restrict__ kW, const float* __restrict__ vW,
    float* __restrict__ out)
{
    const int n    = blockIdx.x;
    const int tid  = threadIdx.x;
    const int wave = tid >> 5;   // wave32
    const int lane = tid & 31;

    // logits / probabilities: [h][w][14]  (first 7 = H side j, last 7 = W side y)
    __shared__ float p[G7 * G7 * 14];

    // ---------------- Phase 1: score GEMMs via WMMA f32 16x16x4 ----------------
    // 14 tasks: task<7 -> H side (w = task), task>=7 -> W side (h = task-7)
    for (int task = wave; task < 14; task += 8) {
        const bool isH = (task < 7);
        const int  idx = isH ? task : task - 7;
        const float* qb = (isH ? qH : qW) + (size_t)(n * G7 + idx) * G7 * CQ; // [row][c]
        const float* kb = (isH ? kH : kW) + (size_t)(n * G7 + idx) * CQ * G7; // [c][col]

        const int  m     = lane & 15;          // A row (h or w)
        const bool mval  = (m < G7);
        const int  mc    = mval ? m : 0;       // clamped -> always in-bounds load
        const bool nval  = mval;               // B col validity (same lane pattern)
        const int  nc    = mc;
        const int  khalf = (lane >> 4) << 1;   // 0 for lanes 0-15, 2 for lanes 16-31

        // Hoist ALL fragment loads: unconditional, so they issue as clauses and
        // overlap; zero-padding applied afterwards with selects (no exec churn).
        float2 af[16], bf[16];
        #pragma unroll
        for (int i = 0; i < 16; ++i) {
            // A: two consecutive K values, 8B-aligned -> global_load_b64
            af[i] = *(const float2*)(qb + mc * CQ + 4 * i + khalf);
        }
        #pragma unroll
        for (int i = 0; i < 16; ++i) {
            bf[i].x = kb[(4 * i + khalf)     * G7 + nc];
            bf[i].y = kb[(4 * i + khalf + 1) * G7 + nc];
        }
        #pragma unroll
        for (int i = 0; i < 16; ++i) {
            af[i].x = mval ? af[i].x : 0.0f;
            af[i].y = mval ? af[i].y : 0.0f;
            bf[i].x = nval ? bf[i].x : 0.0f;
            bf[i].y = nval ? bf[i].y : 0.0f;
        }

        v8f acc = {};
        #pragma unroll
        for (int i = 0; i < 16; ++i) {
            v2f av = {af[i].x, af[i].y};
            v2f bv = {bf[i].x, bf[i].y};
            // D = A(16x4) * B(4x16) + C ; EXEC all-1s here.
            acc = __builtin_amdgcn_wmma_f32_16x16x4_f32(
                false, av, false, bv, (short)0, acc, false, false);
        }

        // C/D layout: lanes 0-15 hold N=lane; VGPR r holds M=r. We need M,N in [0,7).
        if (lane < G7) {
            #pragma unroll
            for (int r = 0; r < G7; ++r) {
                float v = acc[r];
                if (isH) {
                    if (r == lane) v = NEGV;                 // mask diagonal h==j
                    p[(r * G7 + idx) * 14 + lane] = v;       // logits[h=r][w=idx][j=lane]
                } else {
                    p[(idx * G7 + r) * 14 + 7 + lane] = v;   // logits[h=idx][w=r][7+y]
                }
            }
        }
    }
    __syncthreads();

    // ---------------- Softmax over the 14 concatenated logits ----------------
    if (tid < G7 * G7) {
        float* row = &p[tid * 14];
        float mx = row[0];
        #pragma unroll
        for (int i = 1; i < 14; ++i) mx = fmaxf(mx, row[i]);
        float e[14];
        float s = 0.0f;
        #pragma unroll
        for (int i = 0; i < 14; ++i) { e[i] = __expf(row[i] - mx); s += e[i]; }
        const float inv = 1.0f / s;
        #pragma unroll
        for (int i = 0; i < 14; ++i) row[i] = e[i] * inv;
    }
    __syncthreads();

    // ---------------- Phase 2: weighted sum over vH / vW ----------------
    // out[n][c][h][w] = sum_j p[h][w][j]   * vH[(n*7+w)][c][j]
    //                 + sum_y p[h][w][7+y] * vW[(n*7+h)][c][y]
    for (int c = tid; c < CV; c += blockDim.x) {
        // Prefetch next c-iteration's rows (global_prefetch_b8) while we compute.
        const int cn = c + blockDim.x;
        if (cn < CV) {
            #pragma unroll
            for (int w = 0; w < G7; ++w)
                __builtin_prefetch(vH + ((size_t)(n * G7 + w) * CV + cn) * G7, 0, 0);
            #pragma unroll
            for (int h = 0; h < G7; ++h)
                __builtin_prefetch(vW + ((size_t)(n * G7 + h) * CV + cn) * G7, 0, 0);
        }

        float vh[G7][G7];
        #pragma unroll
        for (int w = 0; w < G7; ++w) {
            const float* src = vH + ((size_t)(n * G7 + w) * CV + c) * G7;
            #pragma unroll
            for (int j = 0; j < G7; ++j) vh[w][j] = src[j];
        }
        float* ob = out + ((size_t)n * CV + c) * (G7 * G7);
        #pragma unroll
        for (int h = 0; h < G7; ++h) {
            float vw[G7];
            const float* srcw = vW + ((size_t)(n * G7 + h) * CV + c) * G7;
            #pragma unroll
            for (int y = 0; y < G7; ++y) vw[y] = srcw[y];
            #pragma unroll
            for (int w = 0; w < G7; ++w) {
                const float* pr = &p[(h * G7 + w) * 14];
                float acc = 0.0f;
                #pragma unroll
                for (int j = 0; j < G7; ++j) acc = fmaf(pr[j],     vh[w][j], acc);
                #pragma unroll
                for (int y = 0; y < G7; ++y) acc = fmaf(pr[7 + y], vw[y],    acc);
                ob[h * G7 + w] = acc;
            }
        }
    }
}

extern "C" void kernel_launch(void* const* d_in, const int* in_sizes, int n_in,
                              void* d_out, int out_size, void* d_ws, size_t ws_size,
                              hipStream_t stream)
{
    const float* qH = (const float*)d_in[0];
    const float* kH = (const float*)d_in[1];
    const float* vH = (const float*)d_in[2];
    const float* qW = (const float*)d_in[3];
    const float* kW = (const float*)d_in[4];
    const float* vW = (const float*)d_in[5];
    float* out = (float*)d_out;

    // N from input sizes: qH has N*7*7*64 elements
    const int N = in_sizes[0] / (G7 * G7 * CQ);

    axial_attn_kernel<<<N, 256, 0, stream>>>(qH, kH, vH, qW, kW, vW, out);
}